// RotRNN_17557826306352
// MI455X (gfx1250) — compile-verified
//
#include <hip/hip_runtime.h>
#include <math.h>

typedef __attribute__((ext_vector_type(2))) float v2f;
typedef __attribute__((ext_vector_type(8))) float v8f;

#define T_LEN 8192
#define MD    2048      // model dim == K for both GEMMs
#define NH    64        // heads
#define NS    32        // per-head state dim
#define NPAIR 16        // NS/2
#define GDIM  2048      // NH*NS
#define CHUNK 512
#define NCHUNK (T_LEN / CHUNK)
#define NPAIRS_TOT 1024 // NH * NPAIR
#define LDS_PITCH 20    // floats per row: 80B, 16B-aligned rows, conflict-free

// ---- CDNA5 async global->LDS copy (ASYNCcnt path), 16 bytes per lane ----
__device__ __forceinline__ void async_copy_b128(void* lds_dst, const void* gsrc) {
  // Generic LDS pointer low 32 bits == LDS byte offset (ISA §10.2 aperture map).
  unsigned int lds_off = (unsigned int)(unsigned long long)lds_dst;
  unsigned long long ga = (unsigned long long)gsrc;
  asm volatile("global_load_async_to_lds_b128 %0, %1, off"
               :
               : "v"(lds_off), "v"(ga)
               : "memory");
}
__device__ __forceinline__ void wait_async0() {
  asm volatile("s_wait_asynccnt 0" ::: "memory");
}

// ---------------- per-head scalars: gamma, xi ----------------
__global__ void k_head_setup(const float* __restrict__ B, const float* __restrict__ glog,
                             float* __restrict__ gam, float* __restrict__ xi) {
  __shared__ float red[256];
  const int h = blockIdx.x;
  const float* Bh = B + (size_t)h * NS * MD;
  float s = 0.f;
  for (int idx = threadIdx.x; idx < NS * MD; idx += 256) {
    float v = Bh[idx];
    s += v * v;
  }
  red[threadIdx.x] = s;
  __syncthreads();
  for (int st = 128; st > 0; st >>= 1) {
    if (threadIdx.x < st) red[threadIdx.x] += red[threadIdx.x + st];
    __syncthreads();
  }
  if (threadIdx.x == 0) {
    float g = __expf(-__expf(glog[h]));
    gam[h] = g;
    xi[h] = sqrtf((1.f - g * g) / red[0]);
  }
}

// ---------------- P = expm(M - M^T), scaling & squaring + Taylor ----------------
__global__ void k_expm(const float* __restrict__ M, float* __restrict__ P) {
  __shared__ float A[NS][NS + 1], Tm[NS][NS + 1], Pa[NS][NS + 1];
  __shared__ float rs[NS];
  __shared__ float s_scale;
  __shared__ int s_sq;
  const int h = blockIdx.x;
  const int tid = threadIdx.x;      // 1024 threads
  const int i = tid >> 5, j = tid & 31;
  const float* Mh = M + (size_t)h * NS * NS;
  A[i][j] = Mh[i * NS + j] - Mh[j * NS + i];
  __syncthreads();
  if (j == 0) {
    float s = 0.f;
    for (int m = 0; m < NS; m++) s += fabsf(A[i][m]);
    rs[i] = s;
  }
  __syncthreads();
  if (tid == 0) {
    float mx = 0.f;
    for (int m = 0; m < NS; m++) mx = fmaxf(mx, rs[m]);
    int sq = 0;
    while (mx > 0.25f && sq < 30) { mx *= 0.5f; sq++; }
    s_sq = sq;
    s_scale = exp2f(-(float)sq);
  }
  __syncthreads();
  const float sc = s_scale;
  const int nsq = s_sq;
  A[i][j] *= sc;
  __syncthreads();
  Pa[i][j] = ((i == j) ? 1.f : 0.f) + A[i][j];
  Tm[i][j] = A[i][j];
  __syncthreads();
  for (int k = 2; k <= 12; k++) {
    float s = 0.f;
    for (int m = 0; m < NS; m++) s += Tm[i][m] * A[m][j];
    s /= (float)k;
    __syncthreads();
    Tm[i][j] = s;
    Pa[i][j] += s;
    __syncthreads();
  }
  for (int q = 0; q < nsq; q++) {
    float s = 0.f;
    for (int m = 0; m < NS; m++) s += Pa[i][m] * Pa[m][j];
    __syncthreads();
    Pa[i][j] = s;
    __syncthreads();
  }
  P[(size_t)h * NS * NS + i * NS + j] = Pa[i][j];
}

// ---------------- W[h*32+n][k] = xi_h * sum_m P[h][m][n] * B[h][m][k] ----------------
__global__ void k_make_W(const float* __restrict__ B, const float* __restrict__ P,
                         const float* __restrict__ xi, float* __restrict__ W) {
  __shared__ float Ps[NS * NS];
  const int h = blockIdx.y;
  const int k = blockIdx.x * 256 + threadIdx.x;
  for (int idx = threadIdx.x; idx < NS * NS; idx += 256)
    Ps[idx] = P[(size_t)h * NS * NS + idx];
  __syncthreads();
  const float x = xi[h];
  const float* Bh = B + (size_t)h * NS * MD;
  float b[NS];
  for (int m = 0; m < NS; m++) b[m] = Bh[(size_t)m * MD + k];
  for (int n = 0; n < NS; n++) {
    float s = 0.f;
    for (int m = 0; m < NS; m++) s += Ps[m * NS + n] * b[m];
    W[((size_t)h * NS + n) * MD + k] = x * s;
  }
}

// ---------------- C2[d][h*32+m] = sum_n C[d][h*32+n] * P[h][n][m] ----------------
__global__ void k_make_C2(const float* __restrict__ C, const float* __restrict__ P,
                          float* __restrict__ C2) {
  __shared__ float Ps[NS * NS];
  const int h = blockIdx.y;
  const int d = blockIdx.x * 256 + threadIdx.x;
  for (int idx = threadIdx.x; idx < NS * NS; idx += 256)
    Ps[idx] = P[(size_t)h * NS * NS + idx];
  __syncthreads();
  float c[NS];
  for (int n = 0; n < NS; n++) c[n] = C[(size_t)d * GDIM + h * NS + n];
  for (int m = 0; m < NS; m++) {
    float s = 0.f;
    for (int n = 0; n < NS; n++) s += c[n] * Ps[n * NS + m];
    C2[(size_t)d * GDIM + h * NS + m] = s;
  }
}

// ---------------- fp32 WMMA GEMM: Out[T x 2048] = In[T x 2048] * Wt[2048 x 2048]^T ----
// Block: 256 thr (8 waves, 4x2), 64x64 tile, BK=16, wave tile 16x32 (2 accums).
// K-tiles staged with global_load_async_to_lds_b128 (ASYNCcnt), double-buffered.
template <bool EPI>
__global__ __launch_bounds__(256) void k_gemm_tn(
    const float* __restrict__ In, const float* __restrict__ Wt, float* __restrict__ Out,
    const float* __restrict__ Dvec, const float* __restrict__ U2) {
  __shared__ float As[2][64][LDS_PITCH];
  __shared__ float Bs[2][64][LDS_PITCH];
  const int tid = threadIdx.x;
  const int lane = tid & 31;
  const int wave = tid >> 5;
  const int waveM = wave >> 1;   // 0..3
  const int waveN = wave & 1;    // 0..1
  const int tBase = blockIdx.x * 64;
  const int nBase = blockIdx.y * 64;
  const int lr = tid >> 2;           // 0..63
  const int lc = (tid & 3) << 2;     // 0,4,8,12 (16B-aligned within 80B row)
  const float* gA = In + (size_t)(tBase + lr) * MD + lc;
  const float* gB = Wt + (size_t)(nBase + lr) * MD + lc;
  v8f acc0 = {};
  v8f acc1 = {};
  const int mrow = waveM * 16 + (lane & 15);
  const int ncol = waveN * 32 + (lane & 15);

  // Prologue: stage K-tile 0 into buffer 0 (direct global -> LDS, no VGPR bounce)
  async_copy_b128(&As[0][lr][lc], gA);
  async_copy_b128(&Bs[0][lr][lc], gB);
  wait_async0();
  __syncthreads();

  int cur = 0;
  for (int k0 = 0; k0 < MD; k0 += 16) {
    const int nxt = cur ^ 1;
    if (k0 + 16 < MD) {  // overlap: copy next tile while computing this one
      async_copy_b128(&As[nxt][lr][lc], gA + k0 + 16);
      async_copy_b128(&Bs[nxt][lr][lc], gB + k0 + 16);
    }
#pragma unroll
    for (int kk = 0; kk < 16; kk += 4) {
      const int kr = kk + ((lane >> 4) << 1);  // lanes 0-15: k,k+1 ; lanes 16-31: k+2,k+3
      v2f a, b0, b1;
      a.x = As[cur][mrow][kr];       a.y = As[cur][mrow][kr + 1];
      b0.x = Bs[cur][ncol][kr];      b0.y = Bs[cur][ncol][kr + 1];
      b1.x = Bs[cur][ncol + 16][kr]; b1.y = Bs[cur][ncol + 16][kr + 1];
      acc0 = __builtin_amdgcn_wmma_f32_16x16x4_f32(false, a, false, b0, (short)0, acc0,
                                                   false, false);
      acc1 = __builtin_amdgcn_wmma_f32_16x16x4_f32(false, a, false, b1, (short)0, acc1,
                                                   false, false);
    }
    wait_async0();
    __syncthreads();
    cur = nxt;
  }
  // C/D layout: VGPR r -> M = r (lanes 0-15) / r+8 (lanes 16-31), N = lane&15
  const int colA = nBase + waveN * 32 + (lane & 15);
  const int colB = colA + 16;
  const int rowB = tBase + waveM * 16 + ((lane >> 4) << 3);
#pragma unroll
  for (int r = 0; r < 8; r++) {
    const int row = rowB + r;
    float o0 = acc0[r], o1 = acc1[r];
    if (EPI) {
      o0 += Dvec[colA] * U2[(size_t)row * MD + colA];
      o1 += Dvec[colB] * U2[(size_t)row * MD + colB];
    }
    Out[(size_t)row * MD + colA] = o0;
    Out[(size_t)row * MD + colB] = o1;
  }
}

// ---------------- chunked linear scan: h_t = gamma*R(theta)*h_{t-1} + x_t ----------
__global__ void k_scan_chunk(const float* __restrict__ X, const float* __restrict__ gam,
                             const float* __restrict__ th, float* __restrict__ cf) {
  const int p = blockIdx.x * 256 + threadIdx.x;  // pair 0..1023
  const int h = p >> 4, j = p & 15;
  const float g = gam[h];
  const float t = th[h * NPAIR + j];
  const float ca = g * __cosf(t), sa = g * __sinf(t);
  const int t0 = blockIdx.y * CHUNK;
  const float2* Xp = reinterpret_cast<const float2*>(X) + (size_t)t0 * NPAIRS_TOT + p;
  float a = 0.f, b = 0.f;
  for (int q = 0; q < CHUNK; q++) {
    float2 xv = Xp[(size_t)q * NPAIRS_TOT];
    float an = ca * a - sa * b + xv.x;
    float bn = sa * a + ca * b + xv.y;
    a = an; b = bn;
  }
  reinterpret_cast<float2*>(cf)[(size_t)blockIdx.y * NPAIRS_TOT + p] = make_float2(a, b);
}

__global__ void k_scan_carry(const float* __restrict__ cf, const float* __restrict__ gam,
                             const float* __restrict__ th, float* __restrict__ carry) {
  const int p = threadIdx.x;  // blockDim 1024
  const int h = p >> 4, j = p & 15;
  const float g = gam[h];
  const float t = th[h * NPAIR + j];
  const float gL = powf(g, (float)CHUNK);
  const float ang = t * (float)CHUNK;
  const float cL = gL * cosf(ang), sL = gL * sinf(ang);
  float a = 0.f, b = 0.f;
  const float2* F = reinterpret_cast<const float2*>(cf);
  float2* Cr = reinterpret_cast<float2*>(carry);
  for (int c = 0; c < NCHUNK; c++) {
    Cr[(size_t)c * NPAIRS_TOT + p] = make_float2(a, b);
    float2 f = F[(size_t)c * NPAIRS_TOT + p];
    float an = cL * a - sL * b + f.x;
    float bn = sL * a + cL * b + f.y;
    a = an; b = bn;
  }
}

__global__ void k_scan_apply(float* __restrict__ X, const float* __restrict__ gam,
                             const float* __restrict__ th, const float* __restrict__ carry) {
  const int p = blockIdx.x * 256 + threadIdx.x;
  const int h = p >> 4, j = p & 15;
  const float g = gam[h];
  const float t = th[h * NPAIR + j];
  const float ca = g * __cosf(t), sa = g * __sinf(t);
  const int t0 = blockIdx.y * CHUNK;
  float2* Xp = reinterpret_cast<float2*>(X) + (size_t)t0 * NPAIRS_TOT + p;
  float2 ini = reinterpret_cast<const float2*>(carry)[(size_t)blockIdx.y * NPAIRS_TOT + p];
  float a = ini.x, b = ini.y;
  for (int q = 0; q < CHUNK; q++) {
    float2 xv = Xp[(size_t)q * NPAIRS_TOT];
    float an = ca * a - sa * b + xv.x;
    float bn = sa * a + ca * b + xv.y;
    a = an; b = bn;
    Xp[(size_t)q * NPAIRS_TOT] = make_float2(a, b);  // in-place: X becomes H
  }
}

extern "C" void kernel_launch(void* const* d_in, const int* in_sizes, int n_in,
                              void* d_out, int out_size, void* d_ws, size_t ws_size,
                              hipStream_t stream) {
  (void)in_sizes; (void)n_in; (void)out_size; (void)ws_size;
  const float* U    = (const float*)d_in[0];  // (T, 2048)
  const float* th   = (const float*)d_in[1];  // (64, 16)
  const float* M    = (const float*)d_in[2];  // (64, 32, 32)
  const float* B    = (const float*)d_in[3];  // (64, 32, 2048)
  const float* C    = (const float*)d_in[4];  // (2048, 2048)
  const float* D    = (const float*)d_in[5];  // (2048,)
  const float* glog = (const float*)d_in[6];  // (64, 1)

  float* ws  = (float*)d_ws;
  float* gam = ws;                          // 64
  float* xi  = ws + 64;                     // 64
  float* P   = ws + 128;                    // 64*32*32 = 65536
  float* Wb  = ws + 128 + 65536;            // 2048*2048 (W, then reused as C2)
  float* X   = Wb + (size_t)GDIM * MD;      // 8192*2048 (X, then in-place H)
  float* cf  = X + (size_t)T_LEN * GDIM;    // 16*1024*2
  float* cr  = cf + (size_t)NCHUNK * NPAIRS_TOT * 2;
  float* Y   = (float*)d_out;

  k_head_setup<<<NH, 256, 0, stream>>>(B, glog, gam, xi);
  k_expm<<<NH, 1024, 0, stream>>>(M, P);
  k_make_W<<<dim3(MD / 256, NH), 256, 0, stream>>>(B, P, xi, Wb);
  k_gemm_tn<false><<<dim3(T_LEN / 64, GDIM / 64), 256, 0, stream>>>(U, Wb, X, nullptr, nullptr);
  k_make_C2<<<dim3(MD / 256, NH), 256, 0, stream>>>(C, P, Wb);  // overwrites W with C2
  k_scan_chunk<<<dim3(NPAIRS_TOT / 256, NCHUNK), 256, 0, stream>>>(X, gam, th, cf);
  k_scan_carry<<<1, 1024, 0, stream>>>(cf, gam, th, cr);
  k_scan_apply<<<dim3(NPAIRS_TOT / 256, NCHUNK), 256, 0, stream>>>(X, gam, th, cr);
  k_gemm_tn<true><<<dim3(T_LEN / 64, MD / 64), 256, 0, stream>>>(X, Wb, Y, D, U);
}